// LastMessageAggregator_64707977282175
// MI455X (gfx1250) — compile-verified
//
#include <hip/hip_runtime.h>
#include <stdint.h>

// Problem constants (fixed by the reference): E=1048576 events, N=65536 nodes,
// D=256 f32 features per message row (1 KB per row).
#define N_NODES 65536
#define ROW_F32 256

// ---------------------------------------------------------------------------
// Phase 0: zero the per-node best-key table (u64 per node) in workspace.
// Must run every launch: harness poisons d_ws once and never re-poisons, and
// we must be replay-deterministic.
// ---------------------------------------------------------------------------
__global__ void lma_init_keys(unsigned long long* __restrict__ keys, int n) {
    int i = blockIdx.x * blockDim.x + threadIdx.x;
    if (i < n) keys[i] = 0ull;
}

// ---------------------------------------------------------------------------
// Phase 1: segmented argmax via packed-key atomicMax.
//   key = (ordered_bits(ts) << 32) | ~event_idx
// ordered_bits: monotonic u32 transform of f32 bits (handles negatives too).
// For equal ts, larger ~idx wins under max => smallest event index, matching
// the reference's first-occurrence tie-break.
// Keys table is 512 KB -> resident in the 192 MB L2, so atomic throughput is
// L2-bound, not HBM-bound.
// ---------------------------------------------------------------------------
__global__ void lma_reduce(const int* __restrict__ node_index,
                           const float* __restrict__ timestamp,
                           unsigned long long* __restrict__ keys,
                           int num_events) {
    int i = blockIdx.x * blockDim.x + threadIdx.x;
    const int stride = gridDim.x * blockDim.x;
    for (; i < num_events; i += stride) {
        const uint32_t b = __float_as_uint(timestamp[i]);
        // b >= 0 -> b ^ 0x80000000 ; b < 0 -> ~b   (totally ordered as u32)
        const uint32_t o = b ^ (uint32_t)(((int32_t)b >> 31) | (int32_t)0x80000000);
        const unsigned long long key =
            ((unsigned long long)o << 32) | (uint32_t)(~(uint32_t)i);
        atomicMax(&keys[node_index[i]], key);
    }
}

// ---------------------------------------------------------------------------
// Phase 2: gather. One wave32 per node. Decode the winning event index, then
// move the 1 KB message row with the CDNA5 async LDS-DMA path:
//   2 x global_load_async_to_lds_b128   (32 lanes x 16B x 2 = 1 KB -> LDS)
//   s_wait_asynccnt 0
//   2 x global_store_async_from_lds_b128 (LDS -> output row)
// Data never round-trips through VGPRs; completion tracked via ASYNCcnt.
// 8 waves/block -> 8 KB LDS staging (out of 320 KB/WGP).
// ---------------------------------------------------------------------------
__global__ void lma_gather(const unsigned long long* __restrict__ keys,
                           const float* __restrict__ messages,
                           const float* __restrict__ timestamp,
                           float* __restrict__ out_ids,
                           float* __restrict__ out_msgs,
                           float* __restrict__ out_ts) {
    __shared__ float smem[8][ROW_F32];   // one 1 KB row buffer per wave

    const int wave = threadIdx.x >> 5;
    const int lane = threadIdx.x & 31;
    const int node = blockIdx.x * 8 + wave;   // grid sized exactly N/8

    const unsigned long long key = keys[node];
    const uint32_t ev = ~(uint32_t)key;       // winning event index

    const float* src = messages + (size_t)ev * ROW_F32;
    float* dst       = out_msgs + (size_t)node * ROW_F32;

    // Wave-relative LDS byte offset: low 32 bits of the flat shared address.
    const uint32_t lds0 =
        (uint32_t)(uintptr_t)(&smem[wave][0]) + (uint32_t)(lane * 16);
    const uint64_t g0 = (uint64_t)(uintptr_t)src + (uint64_t)(lane * 16);
    const uint64_t s0 = (uint64_t)(uintptr_t)dst + (uint64_t)(lane * 16);

    // memory -> LDS (two halves of the 1 KB row; offset applies to both sides)
    asm volatile("global_load_async_to_lds_b128 %0, %1, off"
                 :: "v"(lds0), "v"(g0) : "memory");
    asm volatile("global_load_async_to_lds_b128 %0, %1, off offset:512"
                 :: "v"(lds0), "v"(g0) : "memory");
    asm volatile("s_wait_asynccnt 0" ::: "memory");

    // LDS -> output row
    asm volatile("global_store_async_from_lds_b128 %0, %1, off"
                 :: "v"(s0), "v"(lds0) : "memory");
    asm volatile("global_store_async_from_lds_b128 %0, %1, off offset:512"
                 :: "v"(s0), "v"(lds0) : "memory");

    if (lane == 0) {
        out_ids[node] = (float)node;       // node_ids == arange(N), as f32 output
        out_ts[node]  = timestamp[ev];
    }

    // Ensure async stores retire before wave end (s_endpgm also waits-idle).
    asm volatile("s_wait_asynccnt 0" ::: "memory");
}

// ---------------------------------------------------------------------------
// Launch: inputs (dict order) = messages [E*D f32], node_index [E i32],
// timestamp [E f32], num_nodes [1 i32]. Output = node_ids[N] ++
// agg_messages[N*256] ++ agg_timestamps[N], all as f32.
// ---------------------------------------------------------------------------
extern "C" void kernel_launch(void* const* d_in, const int* in_sizes, int n_in,
                              void* d_out, int out_size, void* d_ws, size_t ws_size,
                              hipStream_t stream) {
    const float* messages  = (const float*)d_in[0];
    const int*   node_idx  = (const int*)d_in[1];
    const float* timestamp = (const float*)d_in[2];
    const int E = in_sizes[1];
    const int N = N_NODES;

    unsigned long long* keys = (unsigned long long*)d_ws;   // N * 8 B = 512 KB

    float* out_ids  = (float*)d_out;
    float* out_msgs = out_ids + N;
    float* out_ts   = out_msgs + (size_t)N * ROW_F32;

    lma_init_keys<<<(N + 255) / 256, 256, 0, stream>>>(keys, N);
    lma_reduce<<<2048, 256, 0, stream>>>(node_idx, timestamp, keys, E);
    lma_gather<<<N / 8, 256, 0, stream>>>(keys, messages, timestamp,
                                          out_ids, out_msgs, out_ts);
}